// ProjectionLayer_70325794505024
// MI455X (gfx1250) — compile-verified
//
#include <hip/hip_runtime.h>
#include <hip/hip_bf16.h>
#include <math.h>

// Problem constants (from the reference)
#define GRIDSZ   50
#define G2       (GRIDSZ * GRIDSZ)   // 2500 grid cells
#define NSENS    256                 // sensors
#define TLEN     1024                // time steps
#define BATCH    64
#define BLOCK    256                 // 8 waves on wave32 CDNA5
#define TT       32                  // T-rows staged per workgroup (32 KB tile)

#define HAVE_ASYNC_LDS (__has_builtin(__builtin_amdgcn_global_load_async_to_lds_b128))

typedef int v4i_t __attribute__((ext_vector_type(4)));
typedef __attribute__((address_space(1))) char  GChar;  // global, byte-addressed
typedef __attribute__((address_space(3))) char  LChar;  // LDS, byte-addressed
typedef __attribute__((address_space(1))) v4i_t GVec;   // global b128
typedef __attribute__((address_space(3))) v4i_t LVec;   // LDS b128

// ---------------------------------------------------------------------------
// Kernel 1: nearest[b,g] = argmin_n ( (lx-gx)^2 + (ly-gy)^2 )
// sqrt is monotonic so squared distance preserves the argmin; strict '<'
// keeps the first index on ties, matching jnp.argmin.
// grid = (ceil(G2/BLOCK), B), block = BLOCK
// ---------------------------------------------------------------------------
__global__ void __launch_bounds__(BLOCK)
nearest_kernel(const float* __restrict__ locs, int* __restrict__ nearest) {
    __shared__ float slx[NSENS];
    __shared__ float sly[NSENS];

    const int b   = blockIdx.y;
    const int tid = threadIdx.x;

    const float* lb = locs + (size_t)b * NSENS * 2;
    for (int i = tid; i < NSENS; i += BLOCK) {
        slx[i] = lb[2 * i + 0] * (GRIDSZ * 0.5f) + (GRIDSZ * 0.5f);
        sly[i] = lb[2 * i + 1] * (GRIDSZ * 0.5f) + (GRIDSZ * 0.5f);
    }
    __syncthreads();

    const int g = blockIdx.x * BLOCK + tid;
    if (g < G2) {
        // gridpoint k = (k / G, k % G) per np.mgrid reshape
        const float gx = (float)(g / GRIDSZ);
        const float gy = (float)(g % GRIDSZ);
        float best = INFINITY;
        int   bi   = 0;
#pragma unroll 8
        for (int n = 0; n < NSENS; ++n) {
            const float dx = slx[n] - gx;
            const float dy = sly[n] - gy;
            const float d  = fmaf(dx, dx, dy * dy);
            if (d < best) { best = d; bi = n; }
        }
        nearest[(size_t)b * G2 + g] = bi;
    }
}

// ---------------------------------------------------------------------------
// Kernel 2: out[b, t0+tt, g] = data[b, t0+tt, nearest[b,g]]
// One workgroup per (b, 32-row tile of T). Stages the contiguous 32KB
// data tile into LDS with async global->LDS b128 copies (ASYNCcnt), stages
// nearest[b,:] (10KB) into LDS, then streams coalesced non-temporal stores.
// grid = (TLEN/TT, B), block = BLOCK
// ---------------------------------------------------------------------------
__global__ void __launch_bounds__(BLOCK)
gather_kernel(const float* __restrict__ data,
              const int*   __restrict__ nearest,
              float*       __restrict__ out) {
    __shared__ __align__(16) float tile[TT * NSENS];  // 32 KB
    __shared__ int snear[G2];                         // 10 KB

    const int b   = blockIdx.y;
    const int t0  = blockIdx.x * TT;
    const int tid = threadIdx.x;

    // ---- stage nearest[b, :] into LDS (normal coalesced loads) ----
    const int* nb = nearest + (size_t)b * G2;
    for (int i = tid; i < G2; i += BLOCK) snear[i] = nb[i];

    // ---- stage data[b, t0:t0+TT, :] (contiguous 32 KB) into LDS ----
    const float* gsrc = data + ((size_t)b * TLEN + (size_t)t0) * NSENS;
#if HAVE_ASYNC_LDS
    {
        GChar* gp = (GChar*)(const_cast<float*>(gsrc));
        LChar* lp = (LChar*)(&tile[0]);
        // 32 KB / 16 B = 2048 b128 transfers; 8 per thread
#pragma unroll
        for (int k = 0; k < (TT * NSENS * 4) / (16 * BLOCK); ++k) {
            const int off = (tid + k * BLOCK) * 16;
            __builtin_amdgcn_global_load_async_to_lds_b128(
                (GVec*)(gp + off), (LVec*)(lp + off), 0, 0);
        }
    }
    asm volatile("s_wait_asynccnt 0" ::: "memory");
#else
    {
        const float4* g4 = (const float4*)gsrc;
        float4*       t4 = (float4*)tile;
        for (int k = tid; k < (TT * NSENS) / 4; k += BLOCK) t4[k] = g4[k];
    }
#endif
    __syncthreads();

    // ---- stream the gathered output: coalesced, non-temporal ----
    float* ob = out + ((size_t)b * TLEN + (size_t)t0) * (size_t)G2;
    for (int g = tid; g < G2; g += BLOCK) {
        const int nv = snear[g];
#pragma unroll 8
        for (int tt = 0; tt < TT; ++tt) {
            __builtin_nontemporal_store(tile[tt * NSENS + nv],
                                        &ob[(size_t)tt * G2 + g]);
        }
    }
}

// ---------------------------------------------------------------------------
extern "C" void kernel_launch(void* const* d_in, const int* in_sizes, int n_in,
                              void* d_out, int out_size, void* d_ws, size_t ws_size,
                              hipStream_t stream) {
    (void)in_sizes; (void)n_in; (void)out_size; (void)ws_size;

    const float* data = (const float*)d_in[0];  // [B, T, N] f32
    const float* locs = (const float*)d_in[1];  // [B, N, 2] f32
    float*       out  = (float*)d_out;          // [B, T, G, G] f32
    int*         near = (int*)d_ws;             // [B, G2] scratch (640 KB)

    dim3 grid1((G2 + BLOCK - 1) / BLOCK, BATCH);
    nearest_kernel<<<grid1, BLOCK, 0, stream>>>(locs, near);

    dim3 grid2(TLEN / TT, BATCH);
    gather_kernel<<<grid2, BLOCK, 0, stream>>>(data, near, out);
}